// Markov_fixe_75076028334598
// MI455X (gfx1250) — compile-verified
//
#include <hip/hip_runtime.h>
#include <cstdint>
#include <cstddef>

// "Last previous event" Hawkes increment for MI455X (gfx1250).
//
// out[i] = exp(-(t[i] - t_pad[i, j_last])), j_last = last index with
//          t_pad[i,j] <= t[i];  0 if none.
//
// HBM-bound single pass over 256 MB (~11 us floor at 23.3 TB/s).
// Fast path (L = NCHUNKS*128): one wave per row; all NCHUNKS chunks issued as
// GLOBAL_LOAD_ASYNC_TO_LDS_B128 up front into a contiguous per-wave LDS row
// buffer (the 24-bit IOFFSET advances LDS dest and global src together, so
// all issues share the same two register operands), then consumed in order
// with s_wait_asynccnt <= NCHUNKS-1-K. 8 KB in flight per wave, 64 KB LDS per
// 256-thread block -> 5 blocks / 40 waves per WGP.

#define WAVES_PER_BLOCK 8
#define CHUNK 128                 // floats per wave per chunk (32 lanes x float4)

// ---- compile-time unrolled async issue: offset folds into IOFFSET ----------
template<int K, int NCH>
struct AsyncIssue {
    static __device__ __forceinline__ void run(uint32_t lds, uint64_t ga) {
        asm volatile("global_load_async_to_lds_b128 %0, %1, off offset:%c2"
                     :: "v"(lds), "v"(ga), "n"(K * (CHUNK * 4)) : "memory");
        AsyncIssue<K + 1, NCH>::run(lds, ga);
    }
};
template<int NCH>
struct AsyncIssue<NCH, NCH> {
    static __device__ __forceinline__ void run(uint32_t, uint64_t) {}
};

// ---- compile-time unrolled consume: in-order completion => wait <= NCH-1-K --
template<int K, int NCH>
struct Consume {
    static __device__ __forceinline__ void run(const float4* __restrict__ buf,
                                               int lane, float tq,
                                               int& best_j, float& best_t) {
        asm volatile("s_wait_asynccnt %c0" :: "n"(NCH - 1 - K) : "memory");
        float4 v = buf[K * 32];                  // ds_load_b128, const offset
        const int jb = K * CHUNK + lane * 4;
        if (v.x <= tq) { best_j = jb + 0; best_t = v.x; }
        if (v.y <= tq) { best_j = jb + 1; best_t = v.y; }
        if (v.z <= tq) { best_j = jb + 2; best_t = v.z; }
        if (v.w <= tq) { best_j = jb + 3; best_t = v.w; }
        Consume<K + 1, NCH>::run(buf, lane, tq, best_j, best_t);
    }
};
template<int NCH>
struct Consume<NCH, NCH> {
    static __device__ __forceinline__ void run(const float4*, int, float,
                                               int&, float&) {}
};

__device__ __forceinline__ void wave_reduce_and_store(
    int best_j, float best_t, float tq, int lane, float* out, int row) {
    #pragma unroll
    for (int off = 16; off >= 1; off >>= 1) {
        int   oj = __shfl_xor(best_j, off, 32);
        float ot = __shfl_xor(best_t, off, 32);
        if (oj > best_j) { best_j = oj; best_t = ot; }
    }
    if (lane == 0) {
        out[row] = (best_j >= 0) ? expf(-(tq - best_t)) : 0.0f;
    }
}

// ---- fast path: L == NCHUNKS * 128 -----------------------------------------
template<int NCHUNKS>
__global__ __launch_bounds__(256) void markov_async_kernel(
    const float* __restrict__ t,
    const float* __restrict__ t_pad,
    float* __restrict__ out,
    int N)
{
    // per-wave contiguous row buffer: NCHUNKS * 512 B (8 KB for NCHUNKS=16)
    __shared__ float4 smem[WAVES_PER_BLOCK][NCHUNKS][32];

    const int wave = threadIdx.x >> 5;           // wave32
    const int lane = threadIdx.x & 31;
    const int row  = blockIdx.x * WAVES_PER_BLOCK + wave;
    if (row >= N) return;                        // whole wave exits together

    const int L = NCHUNKS * CHUNK;
    const float tq = t[row];
    const float* __restrict__ rowp = t_pad + (size_t)row * (size_t)L;

    // low 32 bits of the shared flat address == LDS byte address (aperture
    // layout: addr[63:32] selects LDS space, addr[31:0] is the LDS offset)
    const uint32_t lds = (uint32_t)(uintptr_t)&smem[wave][0][lane];
    const uint64_t ga  = (uint64_t)(uintptr_t)(rowp + lane * 4);

    // issue all chunks: NCHUNKS identical instructions, only IOFFSET differs
    AsyncIssue<0, NCHUNKS>::run(lds, ga);

    int   best_j = -1;
    float best_t = 0.0f;
    Consume<0, NCHUNKS>::run(&smem[wave][0][lane], lane, tq, best_j, best_t);

    wave_reduce_and_store(best_j, best_t, tq, lane, out, row);
}

// ---- generic fallback (any L): plain coalesced float4/float loads ----------
__global__ __launch_bounds__(256) void markov_generic_kernel(
    const float* __restrict__ t,
    const float* __restrict__ t_pad,
    float* __restrict__ out,
    int N, int L)
{
    const int wave = threadIdx.x >> 5;
    const int lane = threadIdx.x & 31;
    const int row  = blockIdx.x * WAVES_PER_BLOCK + wave;
    if (row >= N) return;

    const float tq = t[row];
    const float* __restrict__ rowp = t_pad + (size_t)row * (size_t)L;

    int   best_j = -1;
    float best_t = 0.0f;

    const int nvec = L / CHUNK;
    for (int k = 0; k < nvec; ++k) {
        const int jb = k * CHUNK + lane * 4;
        float4 v = *(const float4*)(rowp + jb);
        if (v.x <= tq) { best_j = jb + 0; best_t = v.x; }
        if (v.y <= tq) { best_j = jb + 1; best_t = v.y; }
        if (v.z <= tq) { best_j = jb + 2; best_t = v.z; }
        if (v.w <= tq) { best_j = jb + 3; best_t = v.w; }
    }
    for (int j = nvec * CHUNK + lane; j < L; j += 32) {
        float v = rowp[j];
        if (v <= tq) { best_j = j; best_t = v; }
    }

    wave_reduce_and_store(best_j, best_t, tq, lane, out, row);
}

extern "C" void kernel_launch(void* const* d_in, const int* in_sizes, int n_in,
                              void* d_out, int out_size, void* d_ws, size_t ws_size,
                              hipStream_t stream) {
    // setup_inputs order: 0=src(i32), 1=dst(i32), 2=t(f32,[N]),
    //                     3=x_pad_simu(f32,[N,8]), 4=t_pad(f32,[N,L])
    const float* t     = (const float*)d_in[2];
    const float* t_pad = (const float*)d_in[4];
    float* out = (float*)d_out;

    const int N = in_sizes[2];
    const int L = (N > 0) ? (in_sizes[4] / N) : 0;
    const int blocks = (N + WAVES_PER_BLOCK - 1) / WAVES_PER_BLOCK;
    if (blocks <= 0) return;

    if (L == 16 * CHUNK) {
        hipLaunchKernelGGL((markov_async_kernel<16>),
                           dim3(blocks), dim3(256), 0, stream,
                           t, t_pad, out, N);
    } else {
        hipLaunchKernelGGL(markov_generic_kernel,
                           dim3(blocks), dim3(256), 0, stream,
                           t, t_pad, out, N, L);
    }
}